// _PointnetSAModuleBase_34651796144289
// MI455X (gfx1250) — compile-verified
//
#include <hip/hip_runtime.h>

typedef __attribute__((ext_vector_type(2))) float v2f;
typedef __attribute__((ext_vector_type(8))) float v8f;

#define B_ 16
#define N_ 8192
#define C_ 64
#define S_ 1024
#define K_ 32

// LDS layout for the MLP kernel (floats). Weights stored with row stride 68
// (conflict-free for the WMMA fragment access pattern lane=(k/2)*16+r).
// h-row layout: [f0..f63 | dx dy dz | 0]  (features first -> 16B aligned for
// async b128 copies; W0 columns permuted to match).
#define LSTR 68
#define W0_OFF 0
#define W1_OFF (64 * LSTR)
#define W2_OFF (128 * LSTR)
#define WTOTAL (256 * LSTR)          // 64 + 64 + 128 rows
#define HROW   68
#define SMEM_FLOATS (WTOTAL + 4 * 32 * HROW)

// ---------------------------------------------------------------------------
// Kernel 1: furthest point sampling. One block (1024 threads) per batch.
// All coords resident in LDS (96KB of CDNA5's 320KB/WGP) so the selected
// point broadcasts via a uniform LDS read: 2 barriers per iteration.
// ---------------------------------------------------------------------------
__global__ __launch_bounds__(1024) void fps_kernel(const float* __restrict__ xyz,
                                                   float* __restrict__ newxyz) {
    extern __shared__ float fsm[];               // sx[N] sy[N] sz[N]
    float* sx = fsm;
    float* sy = fsm + N_;
    float* sz = fsm + 2 * N_;
    __shared__ unsigned long long red[32];
    __shared__ int sbest;

    const int b = blockIdx.x;
    const int t = threadIdx.x;
    const int lane = t & 31;
    const int w = t >> 5;
    const float* base = xyz + (size_t)b * N_ * 3;

    float px[8], py[8], pz[8], dd[8];
#pragma unroll
    for (int j = 0; j < 8; ++j) {
        int i = t + j * 1024;
        px[j] = base[i * 3 + 0];
        py[j] = base[i * 3 + 1];
        pz[j] = base[i * 3 + 2];
        dd[j] = 3.402823466e38f;
        sx[i] = px[j]; sy[i] = py[j]; sz[i] = pz[j];
    }
    float* outx = newxyz + (size_t)b * S_ * 3;
    if (t == 0) { outx[0] = px[0]; outx[1] = py[0]; outx[2] = pz[0]; }
    __syncthreads();

    int best = 0;                                // selected[0] = 0
    for (int k = 1; k < S_; ++k) {
        const float lx = sx[best], ly = sy[best], lz = sz[best];
        unsigned long long key = 0ull;
#pragma unroll
        for (int j = 0; j < 8; ++j) {
            float dx = px[j] - lx, dy = py[j] - ly, dz = pz[j] - lz;
            float d = dx * dx + dy * dy + dz * dz;
            dd[j] = fminf(dd[j], d);
            // pack: f32 bits (non-negative -> order preserved), tie -> min index
            unsigned long long kj =
                ((unsigned long long)__float_as_uint(dd[j]) << 32) |
                (unsigned)(N_ - 1 - (t + j * 1024));
            if (kj > key) key = kj;
        }
        for (int off = 16; off > 0; off >>= 1) {
            unsigned long long o = __shfl_down(key, off, 32);
            if (o > key) key = o;
        }
        if (lane == 0) red[w] = key;
        __syncthreads();
        if (w == 0) {
            unsigned long long kk = red[lane];
            for (int off = 16; off > 0; off >>= 1) {
                unsigned long long o = __shfl_down(kk, off, 32);
                if (o > kk) kk = o;
            }
            if (lane == 0) sbest = N_ - 1 - (int)(kk & 0xffffffffu);
        }
        __syncthreads();
        best = sbest;
        if (t == 0) {
            outx[k * 3 + 0] = sx[best];
            outx[k * 3 + 1] = sy[best];
            outx[k * 3 + 2] = sz[best];
        }
    }
}

// ---------------------------------------------------------------------------
// Kernel 2: transpose features (B,C,N) -> (B,N,C) so neighbor gathers are
// contiguous 256B rows.
// ---------------------------------------------------------------------------
__global__ void transpose_kernel(const float* __restrict__ f, float* __restrict__ ft) {
    __shared__ float tile[32][33];
    const int b = blockIdx.z;
    const int n0 = blockIdx.x * 32;
    const int c0 = blockIdx.y * 32;
    for (int i = threadIdx.y; i < 32; i += 8)
        tile[i][threadIdx.x] = f[((size_t)b * C_ + c0 + i) * N_ + n0 + threadIdx.x];
    __syncthreads();
    for (int i = threadIdx.y; i < 32; i += 8)
        ft[((size_t)b * N_ + n0 + i) * C_ + c0 + threadIdx.x] = tile[threadIdx.x][i];
}

// ---------------------------------------------------------------------------
// Kernel 3: ball query. One wave per (b,s) row; ordered scan with ballot +
// prefix popcount reproduces "first NSAMPLE within radius, pad with first".
// ---------------------------------------------------------------------------
__global__ __launch_bounds__(256) void ballquery_kernel(const float* __restrict__ xyz,
                                                        const float* __restrict__ newxyz,
                                                        int* __restrict__ wsidx) {
    const int warp = threadIdx.x >> 5;
    const int lane = threadIdx.x & 31;
    const int row = blockIdx.x * 8 + warp;       // 0 .. B*S-1
    const int b = row >> 10;
    const float* c = newxyz + (size_t)row * 3;
    const float cx = c[0], cy = c[1], cz = c[2];
    const float* px = xyz + (size_t)b * N_ * 3;
    const float r2 = 0.04f;                      // f32(RADIUS*RADIUS)
    int* out = wsidx + (size_t)row * K_;

    int cnt = 0;
    int first = 0;
    for (int basei = 0; basei < N_; basei += 32) {
        int i = basei + lane;
        float dx = px[i * 3 + 0] - cx;
        float dy = px[i * 3 + 1] - cy;
        float dz = px[i * 3 + 2] - cz;
        bool in = (dx * dx + dy * dy + dz * dz) <= r2;
        unsigned m = (unsigned)__ballot(in);
        if (in) {
            int pos = cnt + __popc(m & ((1u << lane) - 1u));
            if (pos < K_) out[pos] = i;
        }
        if (cnt == 0 && m) first = basei + __ffs(m) - 1;
        cnt += __popc(m);
        if (cnt >= K_) break;
    }
    if (cnt < K_) {
        int f = (cnt > 0) ? first : 0;
        if (lane >= cnt) out[lane] = f;          // lanes cnt..31 fill pad slots
    }
}

// ---------------------------------------------------------------------------
// Kernel 4: fused group + 3-layer MLP (WMMA f32 16x16x4) + max over K.
// One wave per center; 4 waves / block; weights + per-wave h-buffer in LDS.
// Feature gather uses CDNA5 async memory->LDS copies (ASYNCcnt).
// A frag: A[m][k] @ lane=(k/2)*16+m, vgpr=k%2.  B frag: B[k][n]=W[n][k] same.
// C frag: vgpr v, lanes 0-15: M=v, N=lane; lanes 16-31: M=v+8.
// ---------------------------------------------------------------------------
__global__ __launch_bounds__(128) void mlp_kernel(
    const float* __restrict__ xyz, const float* __restrict__ featT,
    const float* __restrict__ newxyz, const int* __restrict__ wsidx,
    const float* __restrict__ W0, const float* __restrict__ b0,
    const float* __restrict__ W1, const float* __restrict__ b1,
    const float* __restrict__ W2, const float* __restrict__ b2,
    float* __restrict__ outF) {
    extern __shared__ float smem[];
    const int tid = threadIdx.x;
    const int warp = tid >> 5;
    const int lane = tid & 31;
    const int ln = lane & 15;
    const int hi = lane >> 4;

    // stage weights in LDS (stride-68 rows). W0 columns permuted to the
    // h-row order [feats | coords | pad]: k<64 -> W0 col 3+k, 64..66 -> col
    // k-64, 67 -> 0.
    for (int i = tid; i < 64 * LSTR; i += 128) {
        int r = i / LSTR, c2 = i % LSTR;
        float wv;
        if (c2 < 64)      wv = W0[r * 67 + 3 + c2];
        else if (c2 < 67) wv = W0[r * 67 + (c2 - 64)];
        else              wv = 0.f;
        smem[W0_OFF + i] = wv;
    }
    for (int i = tid; i < 64 * 64; i += 128) {
        int r = i >> 6, c2 = i & 63;
        smem[W1_OFF + r * LSTR + c2] = W1[i];
    }
    for (int i = tid; i < 128 * 64; i += 128) {
        int r = i >> 6, c2 = i & 63;
        smem[W2_OFF + r * LSTR + c2] = W2[i];
    }

    // gather this wave's 32 neighbors: [f0..f63 | dx dy dz | 0]
    const int center = blockIdx.x * 4 + warp;
    const int b = center >> 10;
    const int s = center & 1023;
    float* hb = smem + WTOTAL + warp * (32 * HROW);
    {
        int nb = wsidx[(size_t)center * K_ + lane];
        const float* cc = newxyz + (size_t)center * 3;
        float cx = cc[0], cy = cc[1], cz = cc[2];
        const float* pp = xyz + ((size_t)b * N_ + nb) * 3;
        float* hr = hb + lane * HROW;
        hr[64] = pp[0] - cx; hr[65] = pp[1] - cy; hr[66] = pp[2] - cz;
        hr[67] = 0.f;
        // async copy 256B of features straight into LDS (no VGPR staging).
        // inst_offset applies to both the global and the LDS address.
        const float* gsrc = featT + ((size_t)b * N_ + nb) * C_;
        unsigned lds_off = (unsigned)(size_t)hr;   // low 32 bits = LDS offset
        asm volatile(
            "global_load_async_to_lds_b128 %0, %1, off\n\t"
            "global_load_async_to_lds_b128 %0, %1, off offset:16\n\t"
            "global_load_async_to_lds_b128 %0, %1, off offset:32\n\t"
            "global_load_async_to_lds_b128 %0, %1, off offset:48\n\t"
            "global_load_async_to_lds_b128 %0, %1, off offset:64\n\t"
            "global_load_async_to_lds_b128 %0, %1, off offset:80\n\t"
            "global_load_async_to_lds_b128 %0, %1, off offset:96\n\t"
            "global_load_async_to_lds_b128 %0, %1, off offset:112\n\t"
            "global_load_async_to_lds_b128 %0, %1, off offset:128\n\t"
            "global_load_async_to_lds_b128 %0, %1, off offset:144\n\t"
            "global_load_async_to_lds_b128 %0, %1, off offset:160\n\t"
            "global_load_async_to_lds_b128 %0, %1, off offset:176\n\t"
            "global_load_async_to_lds_b128 %0, %1, off offset:192\n\t"
            "global_load_async_to_lds_b128 %0, %1, off offset:208\n\t"
            "global_load_async_to_lds_b128 %0, %1, off offset:224\n\t"
            "global_load_async_to_lds_b128 %0, %1, off offset:240\n\t"
            "s_wait_asynccnt 0"
            :: "v"(lds_off), "v"(gsrc) : "memory");
    }
    __syncthreads();

    // ---- layer 0: (32x68) x (68x64), then layer 1: (32x64) x (64x64) ----
#pragma unroll 1
    for (int layer = 0; layer < 2; ++layer) {
        const float* wl = smem + (layer == 0 ? W0_OFF : W1_OFF);
        const float* bl = (layer == 0) ? b0 : b1;
        const int kmax = (layer == 0) ? 68 : 64;
        v8f acc[2][4];
#pragma unroll
        for (int mt = 0; mt < 2; ++mt)
#pragma unroll
            for (int nt = 0; nt < 4; ++nt)
                acc[mt][nt] = (v8f){0, 0, 0, 0, 0, 0, 0, 0};
#pragma unroll 1
        for (int kk = 0; kk < kmax; kk += 4) {
            const int kA = kk + 2 * hi;
            v2f a0, a1;
            a0.x = hb[ln * HROW + kA];        a0.y = hb[ln * HROW + kA + 1];
            a1.x = hb[(16 + ln) * HROW + kA]; a1.y = hb[(16 + ln) * HROW + kA + 1];
#pragma unroll
            for (int nt = 0; nt < 4; ++nt) {
                v2f bb;
                bb.x = wl[(nt * 16 + ln) * LSTR + kA];
                bb.y = wl[(nt * 16 + ln) * LSTR + kA + 1];
                acc[0][nt] = __builtin_amdgcn_wmma_f32_16x16x4_f32(
                    false, a0, false, bb, (short)0, acc[0][nt], false, false);
                acc[1][nt] = __builtin_amdgcn_wmma_f32_16x16x4_f32(
                    false, a1, false, bb, (short)0, acc[1][nt], false, false);
            }
        }
        // bias + ReLU, write back into h-buffer (cols 0..63)
#pragma unroll
        for (int nt = 0; nt < 4; ++nt) {
            float bv = bl[nt * 16 + ln];
#pragma unroll
            for (int mt = 0; mt < 2; ++mt)
#pragma unroll
                for (int v = 0; v < 8; ++v) {
                    float x = fmaxf(acc[mt][nt][v] + bv, 0.f);
                    int m = mt * 16 + hi * 8 + v;
                    hb[m * HROW + nt * 16 + ln] = x;
                }
        }
    }

    // ---- layer 2: (32x64) x (64x128) + max over the 32 neighbors ----
    const float* w2 = smem + W2_OFF;
#pragma unroll 1
    for (int nt = 0; nt < 8; ++nt) {
        v8f c0 = (v8f){0, 0, 0, 0, 0, 0, 0, 0};
        v8f c1 = (v8f){0, 0, 0, 0, 0, 0, 0, 0};
#pragma unroll 1
        for (int kk = 0; kk < 64; kk += 4) {
            const int kA = kk + 2 * hi;
            v2f a0, a1, bb;
            a0.x = hb[ln * HROW + kA];        a0.y = hb[ln * HROW + kA + 1];
            a1.x = hb[(16 + ln) * HROW + kA]; a1.y = hb[(16 + ln) * HROW + kA + 1];
            bb.x = w2[(nt * 16 + ln) * LSTR + kA];
            bb.y = w2[(nt * 16 + ln) * LSTR + kA + 1];
            c0 = __builtin_amdgcn_wmma_f32_16x16x4_f32(false, a0, false, bb, (short)0, c0, false, false);
            c1 = __builtin_amdgcn_wmma_f32_16x16x4_f32(false, a1, false, bb, (short)0, c1, false, false);
        }
        // max over M (neighbors): per-lane 16 rows, then swap halves
        float mx = c0[0];
#pragma unroll
        for (int v = 1; v < 8; ++v) mx = fmaxf(mx, c0[v]);
#pragma unroll
        for (int v = 0; v < 8; ++v) mx = fmaxf(mx, c1[v]);
        mx = fmaxf(mx, __shfl_xor(mx, 16, 32));
        float res = fmaxf(mx + b2[nt * 16 + ln], 0.f);
        if (lane < 16)
            outF[((size_t)b * 128 + nt * 16 + ln) * S_ + s] = res;
    }
}

// ---------------------------------------------------------------------------
extern "C" void kernel_launch(void* const* d_in, const int* in_sizes, int n_in,
                              void* d_out, int out_size, void* d_ws, size_t ws_size,
                              hipStream_t stream) {
    const float* xyz  = (const float*)d_in[0];
    const float* feat = (const float*)d_in[1];
    const float* W0   = (const float*)d_in[2];
    const float* b0   = (const float*)d_in[3];
    const float* W1   = (const float*)d_in[4];
    const float* b1   = (const float*)d_in[5];
    const float* W2   = (const float*)d_in[6];
    const float* b2   = (const float*)d_in[7];

    float* out    = (float*)d_out;
    float* newxyz = out;                              // (B,S,3)
    float* outF   = out + (size_t)B_ * S_ * 3;        // (B,128,S)

    int*   wsidx = (int*)d_ws;                                        // B*S*K ints (2 MB)
    float* featT = (float*)((char*)d_ws + (size_t)B_ * S_ * K_ * 4);  // B*N*C floats (32 MB)

    fps_kernel<<<B_, 1024, 3 * N_ * sizeof(float), stream>>>(xyz, newxyz);
    transpose_kernel<<<dim3(N_ / 32, C_ / 32, B_), dim3(32, 8), 0, stream>>>(feat, featT);
    ballquery_kernel<<<(B_ * S_) / 8, 256, 0, stream>>>(xyz, newxyz, wsidx);
    mlp_kernel<<<(B_ * S_) / 4, 128, SMEM_FLOATS * 4, stream>>>(
        xyz, featT, newxyz, wsidx, W0, b0, W1, b1, W2, b2, outF);
}